// RandomFlipLR_71734543777923
// MI455X (gfx1250) — compile-verified
//
#include <hip/hip_runtime.h>

// RandomFlipLR: out[b,c,h,w] = flip_mask[b] ? in[b,c,h,W-1-w] : in[b,c,h,w]
// Pure streaming permutation: 201 MB in + 201 MB out, 0 FLOPs.
// MI455X roofline: 402.6 MB / 23.3 TB/s ~= 17.3 us -> memory-bound.
// Strategy: 128-bit vmem with non-temporal hints, and a *scalar* per-block
// flip branch (rows-per-block divides C*H so batch index is block-uniform).

typedef float v4f __attribute__((ext_vector_type(4)));

constexpr unsigned Bn = 64, Cn = 3, Hn = 512, Wn = 512;
constexpr unsigned W4   = Wn / 4;          // 128 float4 groups per row
constexpr unsigned ROWS = Bn * Cn * Hn;    // 98304 rows of (b,c,h)
constexpr unsigned RPB  = 8;               // rows per block; 1536 % 8 == 0
                                           // -> batch index b uniform per block
constexpr unsigned ITER = RPB / 2;         // 256 threads cover 2 rows/iter

__global__ __launch_bounds__(256) void RandomFlipLR_kernel(
    const v4f* __restrict__ in, const int* __restrict__ flip_mask,
    v4f* __restrict__ out) {
  const unsigned row0 = blockIdx.x * RPB;          // scalar
  const unsigned b    = row0 / (Cn * Hn);          // scalar (SALU magic-mul)
  const unsigned tid  = threadIdx.x;
  const unsigned t    = tid & (W4 - 1);            // column group within row
  const unsigned rsub = tid >> 7;                  // 0 or 1: row within pair

  // Force the flip predicate into an SGPR -> s_cmp + s_cbranch, no EXEC games.
  const int mflip = __builtin_amdgcn_readfirstlane(flip_mask[b]);

  if (mflip > 0) {
    // Mirrored copy. Each row is still one contiguous, aligned 2KB span of
    // b128 accesses (reverse lane order coalesces identically).
    v4f v[ITER];
#pragma unroll
    for (unsigned i = 0; i < ITER; ++i) {
      const unsigned base = (row0 + 2u * i + rsub) << 7;
      v[i] = __builtin_nontemporal_load(in + base + (W4 - 1u - t));
    }
#pragma unroll
    for (unsigned i = 0; i < ITER; ++i) {
      const unsigned base = (row0 + 2u * i + rsub) << 7;
      v4f r = __builtin_shufflevector(v[i], v[i], 3, 2, 1, 0);
      __builtin_nontemporal_store(r, out + base + t);
    }
  } else {
    // Straight copy: fully linear, 4KB (256-float4) block stride per iteration.
    const unsigned blockBase = row0 << 7;            // row0 * W4
    v4f v[ITER];
#pragma unroll
    for (unsigned i = 0; i < ITER; ++i)
      v[i] = __builtin_nontemporal_load(in + blockBase + i * 256u + tid);
#pragma unroll
    for (unsigned i = 0; i < ITER; ++i)
      __builtin_nontemporal_store(v[i], out + blockBase + i * 256u + tid);
  }
}

extern "C" void kernel_launch(void* const* d_in, const int* in_sizes, int n_in,
                              void* d_out, int out_size, void* d_ws, size_t ws_size,
                              hipStream_t stream) {
  (void)in_sizes; (void)n_in; (void)out_size; (void)d_ws; (void)ws_size;
  const v4f* in        = (const v4f*)d_in[0];
  const int* flip_mask = (const int*)d_in[1];
  v4f*       out       = (v4f*)d_out;

  dim3 grid(ROWS / RPB); // 12288 blocks x 8 wave32, 4x b128 ld+st per thread
  RandomFlipLR_kernel<<<grid, dim3(256), 0, stream>>>(in, flip_mask, out);
}